// CapsuleLayer_51488067944651
// MI455X (gfx1250) — compile-verified
//
#include <hip/hip_runtime.h>

// CDNA5 / gfx1250 fused capsule-routing kernel, round 2.
// One workgroup = (class c, tile of 16 batch rows). u_hat tiles recomputed on
// demand with V_WMMA_F32_16X16X4_F32 (f32 precision, K=8 as 2 chained K=4
// WMMAs). Routing logits AND precomputed softmax weights live in LDS in
// [r][b] layout so the hot loop reads weights as two ds_load_b128 per lane
// (no v_exp in the WMMA loop).

typedef float v2f __attribute__((ext_vector_type(2)));
typedef float v4f __attribute__((ext_vector_type(4)));
typedef float v8f __attribute__((ext_vector_type(8)));

#define RNODES 1152
#define NWAVES 8
#define RPW    (RNODES / NWAVES)   // 144 route nodes per wave
#define NITER  3
#define CAPS_EPS 1e-7f

__device__ __forceinline__ float redsum16(float v) {
  // sum across the 16-lane half (masks 1,2,4,8 never cross the half boundary)
  v += __shfl_xor(v, 1);
  v += __shfl_xor(v, 2);
  v += __shfl_xor(v, 4);
  v += __shfl_xor(v, 8);
  return v;
}
__device__ __forceinline__ float redmax16(float v) {
  v = fmaxf(v, __shfl_xor(v, 1));
  v = fmaxf(v, __shfl_xor(v, 2));
  v = fmaxf(v, __shfl_xor(v, 4));
  v = fmaxf(v, __shfl_xor(v, 8));
  return v;
}

// Compute the 16x16 (batch x out) u_hat tile for route node r with f32 WMMA.
// A layout (32-bit A 16x4): lanes 0-15 -> K=0,1 ; lanes 16-31 -> K=2,3.
// B layout mirrored: lane = N, VGPR = K within the half's K pair.
__device__ __forceinline__ v8f uhat_tile(const float* __restrict__ xrow,
                                         const float* __restrict__ wrow,
                                         int hi, int lo) {
  const float* xp = xrow + 2 * hi;            // A[m][2hi], A[m][2hi+1]
  v2f a0 = { xp[0], xp[1] };                  // i = 2hi, 2hi+1
  v2f a1 = { xp[4], xp[5] };                  // i = 4+2hi, 4+2hi+1
  const float* wp = wrow + (2 * hi) * 16 + lo;
  v2f b0 = { wp[0],  wp[16] };                // B[2hi][n], B[2hi+1][n]
  v2f b1 = { wp[64], wp[80] };                // same with i += 4
  v8f u = {0.f, 0.f, 0.f, 0.f, 0.f, 0.f, 0.f, 0.f};
  u = __builtin_amdgcn_wmma_f32_16x16x4_f32(false, a0, false, b0,
                                            (short)0, u, false, false);
  u = __builtin_amdgcn_wmma_f32_16x16x4_f32(false, a1, false, b1,
                                            (short)0, u, false, false);
  return u;
}

__global__ __launch_bounds__(256) void capsule_route_kernel(
    const float* __restrict__ x,   // [256][1152][8]
    const float* __restrict__ W,   // [10][1152][8][16]
    float* __restrict__ out)       // [10][256][16]
{
  // [r][b_local] layout: per-lane weight reads are contiguous (b128 pairs)
  __shared__ __align__(16) float blog[RNODES * 16];   // routing logits (73.7 KB)
  __shared__ __align__(16) float wexp[RNODES * 16];   // exp(blog - max) (73.7 KB)
  __shared__ float s_part[NWAVES * 256];              // per-wave partial s[b][o]
  __shared__ float vout_lds[256];                     // squashed output v[b][o]
  __shared__ float bz[16];

  const int tid  = threadIdx.x;
  const int wave = tid >> 5;
  const int lane = tid & 31;
  const int hi   = lane >> 4;     // selects K-pair / M-half
  const int lo   = lane & 15;     // M for A-frag, N for B/D-frag
  const int c    = blockIdx.x >> 4;          // 0..9
  const int b0   = (blockIdx.x & 15) << 4;   // batch tile start

  // zero logits
  for (int i = tid; i < RNODES * 16; i += 256) blog[i] = 0.f;
  __syncthreads();

  const int r_begin = wave * RPW;
  const int r_end   = r_begin + RPW;
  const float* xbase = x + (size_t)(b0 + lo) * (RNODES * 8);
  const float* wbase = W + (size_t)c * (RNODES * 8 * 16);

  for (int it = 0; it < NITER; ++it) {
    // ---- softmax over R per batch row; store exp(blog - max) into wexp ----
    {
      const int g = tid >> 4, sub = tid & 15;   // 16 threads per batch row
      float m = -3.4e38f;
      for (int r = sub; r < RNODES; r += 16)
        m = fmaxf(m, blog[r * 16 + g]);
      m = redmax16(m);
      float z = 0.f;
      for (int r = sub; r < RNODES; r += 16) {
        float e = __expf(blog[r * 16 + g] - m);
        wexp[r * 16 + g] = e;
        z += e;
      }
      z = redsum16(z);
      if (sub == 0) bz[g] = z;
    }
    __syncthreads();

    // ---- pass S: s[b][o] = sum_r wexp[r][b] * u_hat[b][r][o] ----
    float sacc[8];
#pragma unroll
    for (int v = 0; v < 8; ++v) sacc[v] = 0.f;

    for (int r = r_begin; r < r_end; ++r) {
      v8f u = uhat_tile(xbase + r * 8, wbase + r * (8 * 16), hi, lo);
      // 8 contiguous weights for this lane's M-half: two ds_load_b128
      v4f wa = *(const v4f*)&wexp[r * 16 + 8 * hi];
      v4f wb = *(const v4f*)&wexp[r * 16 + 8 * hi + 4];
#pragma unroll
      for (int v = 0; v < 4; ++v) sacc[v]     = fmaf(wa[v], u[v],     sacc[v]);
#pragma unroll
      for (int v = 0; v < 4; ++v) sacc[v + 4] = fmaf(wb[v], u[v + 4], sacc[v + 4]);
    }
#pragma unroll
    for (int v = 0; v < 8; ++v)
      s_part[wave * 256 + (v + 8 * hi) * 16 + lo] = sacc[v];
    __syncthreads();

    // ---- squash over O (deterministic cross-wave reduce) ----
    {
      const int b = tid >> 4;                  // o = tid & 15
      float val = 0.f;
#pragma unroll
      for (int w = 0; w < NWAVES; ++w) val += s_part[w * 256 + tid];
      val /= bz[b];
      float sn = redsum16(val * val);
      float scale = (sn / (1.f + sn)) * rsqrtf(sn + CAPS_EPS);
      float vo = scale * val;
      vout_lds[tid] = vo;
      if (it == NITER - 1)
        out[(size_t)(c * 256 + b0 + b) * 16 + (tid & 15)] = vo;
    }
    __syncthreads();

    if (it == NITER - 1) break;

    // ---- pass B: blog[r][b] += sum_o u_hat[b][r][o] * v[b][o] ----
    float vo8[8];
#pragma unroll
    for (int v = 0; v < 8; ++v) vo8[v] = vout_lds[(v + 8 * hi) * 16 + lo];

    for (int r = r_begin; r < r_end; ++r) {
      v8f u = uhat_tile(xbase + r * 8, wbase + r * (8 * 16), hi, lo);
#pragma unroll
      for (int v = 0; v < 8; ++v) {
        float t = redsum16(u[v] * vo8[v]);     // dot over the 16 o-lanes
        if (lo == 0) blog[r * 16 + 8 * hi + v] += t;  // unique writer
      }
    }
    __syncthreads();
  }
}

extern "C" void kernel_launch(void* const* d_in, const int* in_sizes, int n_in,
                              void* d_out, int out_size, void* d_ws, size_t ws_size,
                              hipStream_t stream) {
  (void)in_sizes; (void)n_in; (void)d_ws; (void)ws_size; (void)out_size;
  const float* x = (const float*)d_in[0];   // [256,1152,8]
  const float* W = (const float*)d_in[1];   // [10,1152,8,16]
  float* out = (float*)d_out;               // [10,256,1,1,16]
  capsule_route_kernel<<<dim3(10 * 16), dim3(256), 0, stream>>>(x, W, out);
}